// MyMultimodal_64020782515011
// MI455X (gfx1250) — compile-verified
//
#include <hip/hip_runtime.h>
#include <hip/hip_bf16.h>
#include <math.h>

#define NN      4096
#define NFEAT   128
#define NHID    64
#define NCLASS  6
#define NHEADS  4
#define WIN     10
#define ALPHA   0.2f
#define NEGINF  (-9.0e15f)

typedef __attribute__((ext_vector_type(2))) float v2f;
typedef __attribute__((ext_vector_type(8))) float v8f;

// ---------------------------------------------------------------------------
// Kernel 1: layer-1 multi-head GEMM  Wh[h] = x @ W_heads[h]
// grid = 256 (16-row stripes), block = 128 (4 waves, wave w == head w)
// f32 WMMA 16x16x4: A 16x4 (2 VGPR), B 4x16 (2 VGPR), C/D 16x16 (8 VGPR)
// x tile staged into LDS with CDNA5 async global->LDS copies (ASYNCcnt).
// ---------------------------------------------------------------------------
__global__ void gat1_gemm(const float* __restrict__ x,
                          const float* __restrict__ W_heads,
                          float* __restrict__ whL1) {
    __shared__ float xs[16][NFEAT + 4];
    const int t = threadIdx.x;
    const int m0 = blockIdx.x * 16;

    // async copy of the 16x128 x tile: 16B per lane per iter, 4 iters
    {
        const unsigned lds_base = (unsigned)(uintptr_t)(&xs[0][0]);
        #pragma unroll
        for (int i = t; i < 16 * NFEAT / 4; i += 128) {
            const int row = i >> 5;              // 32 float4 per row
            const int col = (i & 31) << 2;
            const float* gsrc = x + (size_t)(m0 + row) * NFEAT + col;
            const unsigned ldst =
                lds_base + (unsigned)((row * (NFEAT + 4) + col) * sizeof(float));
            asm volatile("global_load_async_to_lds_b128 %0, %1, off"
                         :: "v"(ldst), "v"(gsrc) : "memory");
        }
        asm volatile("s_wait_asynccnt 0x0" ::: "memory");
    }
    __syncthreads();

    const int wave = t >> 5;             // head index
    const int lane = t & 31;
    const int row16 = lane & 15;         // A: M row / B: N col within tile
    const int khalf = (lane >> 4) << 1;  // 0 for lanes 0-15, 2 for lanes 16-31
    const float* Wb = W_heads + (size_t)wave * NFEAT * NHID;   // [128][64]
    float* outp = whL1 + (size_t)wave * NN * NHID;

    for (int nt = 0; nt < 4; ++nt) {
        const int col = nt * 16 + row16;
        v8f acc = {};
        #pragma unroll 4
        for (int k = 0; k < NFEAT; k += 4) {
            v2f a, b;
            a.x = xs[row16][k + khalf];
            a.y = xs[row16][k + khalf + 1];
            b.x = Wb[(size_t)(k + khalf) * NHID + col];
            b.y = Wb[(size_t)(k + khalf + 1) * NHID + col];
            acc = __builtin_amdgcn_wmma_f32_16x16x4_f32(
                false, a, false, b, (short)0, acc, false, false);
        }
        const int rbase = (lane < 16) ? 0 : 8;
        #pragma unroll
        for (int r = 0; r < 8; ++r)
            outp[(size_t)(m0 + rbase + r) * NHID + col] = acc[r];
    }
}

// ---------------------------------------------------------------------------
// Kernel 2: f1[h][i] = Wh[h][i] . a1[h],  f2[h][i] = Wh[h][i] . a2[h]
// ---------------------------------------------------------------------------
__global__ void gat1_f(const float* __restrict__ whL1,
                       const float* __restrict__ a_heads,
                       float* __restrict__ f1, float* __restrict__ f2) {
    const int idx = blockIdx.x * blockDim.x + threadIdx.x;   // h*N + i
    if (idx >= NHEADS * NN) return;
    const int h = idx / NN;
    const float* wh = whL1 + (size_t)idx * NHID;
    const float* a1 = a_heads + (size_t)h * 2 * NHID;
    const float* a2 = a1 + NHID;
    float s1 = 0.f, s2 = 0.f;
    #pragma unroll 8
    for (int c = 0; c < NHID; ++c) {
        const float v = wh[c];
        s1 += v * a1[c];
        s2 += v * a2[c];
    }
    f1[idx] = s1; f2[idx] = s2;
}

// ---------------------------------------------------------------------------
// Kernel 3: banded attention + ELU for layer 1. One wave per (node, head).
// block = 256 (8 waves), grid = N*NHEADS/8
// ---------------------------------------------------------------------------
__global__ void gat1_attn(const float* __restrict__ whL1,
                          const float* __restrict__ f1,
                          const float* __restrict__ f2,
                          const float* __restrict__ adj,
                          float* __restrict__ h1) {
    const int gw = (blockIdx.x * blockDim.x + threadIdx.x) >> 5;
    const int lane = threadIdx.x & 31;
    if (gw >= NN * NHEADS) return;
    const int i = gw >> 2;
    const int h = gw & 3;

    const int jlo = (i - WIN > 0) ? i - WIN : 0;
    const int jhi = (i + WIN < NN - 1) ? i + WIN : NN - 1;
    const int nj = jhi - jlo + 1;                 // <= 21

    const float fi = f1[(size_t)h * NN + i];
    float my_e = -3.0e38f;
    if (lane < nj) {
        const int j = jlo + lane;
        float ev = fi + f2[(size_t)h * NN + j];
        ev = (ev > 0.f) ? ev : ALPHA * ev;        // LeakyReLU
        if (!(adj[(size_t)i * NN + j] > 0.f)) ev = NEGINF;
        my_e = ev;
    }
    // wave32 max-reduce
    float m = my_e;
    #pragma unroll
    for (int off = 16; off > 0; off >>= 1) m = fmaxf(m, __shfl_xor(m, off, 32));
    float w = (lane < nj) ? __expf(my_e - m) : 0.f;
    float s = w;
    #pragma unroll
    for (int off = 16; off > 0; off >>= 1) s += __shfl_xor(s, off, 32);
    const float inv = 1.f / s;

    // each lane owns output cols lane and lane+32
    const float* whh = whL1 + (size_t)h * NN * NHID;
    float acc0 = 0.f, acc1 = 0.f;
    #pragma unroll
    for (int jj = 0; jj < 2 * WIN + 1; ++jj) {
        if (jj >= nj) break;
        const float att = __shfl(w, jj, 32) * inv;
        const float* p = whh + (size_t)(jlo + jj) * NHID;
        acc0 += att * p[lane];
        acc1 += att * p[lane + 32];
    }
    acc0 = (acc0 > 0.f) ? acc0 : (__expf(acc0) - 1.f);   // ELU
    acc1 = (acc1 > 0.f) ? acc1 : (__expf(acc1) - 1.f);
    float* dst = h1 + (size_t)i * (NHEADS * NHID) + h * NHID;
    dst[lane] = acc0;
    dst[lane + 32] = acc1;
}

// ---------------------------------------------------------------------------
// Kernel 4: layer-2 GEMM  Wh2 = h_cat @ W_out  (6 cols, zero-padded to 16)
// grid = 64 (64-row super-tiles), block = 128 (4 waves, 16 rows each)
// ---------------------------------------------------------------------------
__global__ void gat2_gemm(const float* __restrict__ h1,
                          const float* __restrict__ W_out,
                          float* __restrict__ wh2) {
    __shared__ float wpad[2 * NFEAT][16];     // 256 x 16 padded weights
    const int t = threadIdx.x;
    for (int idx = t; idx < 2 * NFEAT * 16; idx += blockDim.x) {
        const int r = idx >> 4, c = idx & 15;
        wpad[r][c] = (c < NCLASS) ? W_out[(size_t)r * NCLASS + c] : 0.f;
    }
    __syncthreads();

    const int wave = t >> 5, lane = t & 31;
    const int m0 = (blockIdx.x * 4 + wave) * 16;
    const int row16 = lane & 15;
    const int khalf = (lane >> 4) << 1;

    v8f acc = {};
    const float* ar = h1 + (size_t)(m0 + row16) * (2 * NFEAT);
    #pragma unroll 4
    for (int k = 0; k < 2 * NFEAT; k += 4) {
        v2f a, b;
        a.x = ar[k + khalf];
        a.y = ar[k + khalf + 1];
        b.x = wpad[k + khalf][row16];
        b.y = wpad[k + khalf + 1][row16];
        acc = __builtin_amdgcn_wmma_f32_16x16x4_f32(
            false, a, false, b, (short)0, acc, false, false);
    }
    if (row16 < NCLASS) {
        const int rbase = (lane < 16) ? 0 : 8;
        #pragma unroll
        for (int r = 0; r < 8; ++r)
            wh2[(size_t)(m0 + rbase + r) * 8 + row16] = acc[r];  // stride-8 pad
    }
}

// ---------------------------------------------------------------------------
// Kernel 5: f1o/f2o = Wh2 . a_out halves (dot-6 per node)
// ---------------------------------------------------------------------------
__global__ void gat2_f(const float* __restrict__ wh2,
                       const float* __restrict__ a_out,
                       float* __restrict__ f1o, float* __restrict__ f2o) {
    const int i = blockIdx.x * blockDim.x + threadIdx.x;
    if (i >= NN) return;
    const float* w = wh2 + (size_t)i * 8;
    float s1 = 0.f, s2 = 0.f;
    #pragma unroll
    for (int c = 0; c < NCLASS; ++c) {
        s1 += w[c] * a_out[c];
        s2 += w[c] * a_out[NCLASS + c];
    }
    f1o[i] = s1; f2o[i] = s2;
}

// ---------------------------------------------------------------------------
// Kernel 6: layer-2 banded attention + ELU + log_softmax. One thread per node.
// ---------------------------------------------------------------------------
__global__ void gat2_attn(const float* __restrict__ wh2,
                          const float* __restrict__ f1o,
                          const float* __restrict__ f2o,
                          const float* __restrict__ adj,
                          float* __restrict__ out) {
    const int i = blockIdx.x * blockDim.x + threadIdx.x;
    if (i >= NN) return;
    const int jlo = (i - WIN > 0) ? i - WIN : 0;
    const int jhi = (i + WIN < NN - 1) ? i + WIN : NN - 1;
    const int nj = jhi - jlo + 1;
    const float fi = f1o[i];

    float ev[2 * WIN + 1];
    float m = -3.0e38f;
    #pragma unroll
    for (int jj = 0; jj < 2 * WIN + 1; ++jj) {
        if (jj >= nj) break;
        const int j = jlo + jj;
        float e = fi + f2o[j];
        e = (e > 0.f) ? e : ALPHA * e;
        if (!(adj[(size_t)i * NN + j] > 0.f)) e = NEGINF;
        ev[jj] = e;
        m = fmaxf(m, e);
    }
    float s = 0.f;
    #pragma unroll
    for (int jj = 0; jj < 2 * WIN + 1; ++jj) {
        if (jj >= nj) break;
        const float w = __expf(ev[jj] - m);
        ev[jj] = w;
        s += w;
    }
    const float inv = 1.f / s;

    float acc[NCLASS];
    #pragma unroll
    for (int c = 0; c < NCLASS; ++c) acc[c] = 0.f;
    #pragma unroll
    for (int jj = 0; jj < 2 * WIN + 1; ++jj) {
        if (jj >= nj) break;
        const float w = ev[jj] * inv;
        const float* p = wh2 + (size_t)(jlo + jj) * 8;
        #pragma unroll
        for (int c = 0; c < NCLASS; ++c) acc[c] += w * p[c];
    }
    // ELU then log_softmax over 6 classes
    float mx = -3.0e38f;
    #pragma unroll
    for (int c = 0; c < NCLASS; ++c) {
        acc[c] = (acc[c] > 0.f) ? acc[c] : (__expf(acc[c]) - 1.f);
        mx = fmaxf(mx, acc[c]);
    }
    float lse = 0.f;
    #pragma unroll
    for (int c = 0; c < NCLASS; ++c) lse += __expf(acc[c] - mx);
    lse = __logf(lse);
    #pragma unroll
    for (int c = 0; c < NCLASS; ++c)
        out[(size_t)i * NCLASS + c] = acc[c] - mx - lse;
}

// ---------------------------------------------------------------------------
extern "C" void kernel_launch(void* const* d_in, const int* in_sizes, int n_in,
                              void* d_out, int out_size, void* d_ws, size_t ws_size,
                              hipStream_t stream) {
    const float* x       = (const float*)d_in[0];   // [4096,128]
    const float* adj     = (const float*)d_in[1];   // [4096,4096]
    const float* W_heads = (const float*)d_in[2];   // [4,128,64]
    const float* a_heads = (const float*)d_in[3];   // [4,128,1]
    const float* W_out   = (const float*)d_in[4];   // [256,6]
    const float* a_out   = (const float*)d_in[5];   // [12,1]
    float* out = (float*)d_out;                     // [4096,6]

    float* ws   = (float*)d_ws;
    float* whL1 = ws;                               // 4*4096*64   = 1048576
    float* f1   = whL1 + (size_t)NHEADS * NN * NHID;        // 16384
    float* f2   = f1 + (size_t)NHEADS * NN;                 // 16384
    float* h1   = f2 + (size_t)NHEADS * NN;                 // 4096*256 = 1048576
    float* wh2  = h1 + (size_t)NN * NHEADS * NHID;          // 4096*8   = 32768
    float* f1o  = wh2 + (size_t)NN * 8;                     // 4096
    float* f2o  = f1o + NN;                                 // 4096
    (void)in_sizes; (void)n_in; (void)out_size; (void)ws_size;

    // layer 1
    gat1_gemm<<<NN / 16, 128, 0, stream>>>(x, W_heads, whL1);
    gat1_f<<<(NHEADS * NN + 255) / 256, 256, 0, stream>>>(whL1, a_heads, f1, f2);
    gat1_attn<<<(NN * NHEADS * 32) / 256, 256, 0, stream>>>(whL1, f1, f2, adj, h1);
    // layer 2
    gat2_gemm<<<NN / 64, 128, 0, stream>>>(h1, W_out, wh2);
    gat2_f<<<(NN + 255) / 256, 256, 0, stream>>>(wh2, a_out, f1o, f2o);
    gat2_attn<<<(NN + 255) / 256, 256, 0, stream>>>(wh2, f1o, f2o, adj, out);
}